// CrossPatchContextModule_20830591386226
// MI455X (gfx1250) — compile-verified
//
#include <hip/hip_runtime.h>
#include <hip/hip_bf16.h>
#include <math.h>

// Problem constants (match reference: B=8, N=256, D=256)
constexpr int kB = 8;
constexpr int kN = 256;
constexpr int kD = 256;
constexpr int kRows = kB * kN;      // 2048 flattened rows
constexpr float kLnEps = 1e-5f;

typedef __attribute__((ext_vector_type(2))) float v2f;
typedef __attribute__((ext_vector_type(8))) float v8f;

// CDNA5 async copy: global -> LDS, 16B per lane, tracked by ASYNCcnt.
__device__ __forceinline__ void async_load_b128(uint32_t lds_addr, const float* gaddr) {
  asm volatile("global_load_async_to_lds_b128 %0, %1, off"
               :: "v"(lds_addr), "v"(gaddr)
               : "memory");
}

// ---------------------------------------------------------------------------
// Kernel 1: hi = x @ W1[:D] + b1 ; hj = x @ W1[D:]
// One wave per 16x64 output strip: 4 WMMA accumulators share one A fragment
// per k-step (loads/WMMA drops 4 -> 2.5). 1024 waves -> 128 blocks.
// ---------------------------------------------------------------------------
__global__ void k1_hproj(const float* __restrict__ x, const float* __restrict__ W1,
                         const float* __restrict__ b1, float* __restrict__ hi,
                         float* __restrict__ hj) {
  const int wid  = blockIdx.x * (blockDim.x >> 5) + (threadIdx.x >> 5);
  const int lane = threadIdx.x & 31;
  const int l16  = lane & 15;
  const int half = lane >> 4;
  const int which = wid >> 9;            // 0 = hi (W1 top), 1 = hj (W1 bottom)
  const int t   = wid & 511;
  const int mt  = t >> 2;                // 128 row tiles
  const int nt4 = t & 3;                 // 4 strips of 64 cols

  const float* A  = x + mt * 16 * kD;
  const float* Bm = W1 + which * kD * kD + nt4 * 64;
  float* out = (which == 0 ? hi : hj) + mt * 16 * kD + nt4 * 64;

  v8f acc[4] = {};
  for (int k = 0; k < kD; k += 4) {
    v2f a;
    const float* ap = A + l16 * kD + k + half * 2;
    a.x = ap[0]; a.y = ap[1];
    const float* brow = Bm + (k + half * 2) * kD + l16;
#pragma unroll
    for (int j = 0; j < 4; ++j) {
      v2f b;
      b.x = brow[j * 16];
      b.y = brow[j * 16 + kD];
      acc[j] = __builtin_amdgcn_wmma_f32_16x16x4_f32(false, a, false, b,
                                                     (short)0, acc[j], false, false);
    }
  }
#pragma unroll
  for (int j = 0; j < 4; ++j) {
    const float bias = (which == 0) ? b1[nt4 * 64 + j * 16 + l16] : 0.0f;
#pragma unroll
    for (int v = 0; v < 8; ++v)
      out[(v + half * 8) * kD + j * 16 + l16] = acc[j][v] + bias;
  }
}

// ---------------------------------------------------------------------------
// Kernel 2: scores[b,n,m] = sum_d relu(hi[b,n,d]+hj[b,m,d]) * W2[d] + b2,
// diagonal forced to 0, then softmax over m. One block per (b,n) row.
// hj is streamed through LDS with double-buffered async-to-LDS copies
// (ASYNCcnt pipeline hides L2 latency behind the ReLU-dot VALU work).
// ---------------------------------------------------------------------------
__global__ void k2_scores_softmax(const float* __restrict__ hi,
                                  const float* __restrict__ hj,
                                  const float* __restrict__ W2,
                                  const float* __restrict__ b2,
                                  float* __restrict__ wout) {
  __shared__ float hi_s[kD];
  __shared__ float w2_s[kD];
  __shared__ float sc_s[kN];
  __shared__ float red_s[16];
  __shared__ float hjbuf[2][8][kD];      // double-buffered 8-row hj tile (2x8KB)

  const int t    = threadIdx.x;
  const int bidx = blockIdx.x;           // flattened b*N + n
  const int n    = bidx & (kN - 1);
  hi_s[t] = hi[bidx * kD + t];
  w2_s[t] = W2[t];

  const int wave = t >> 5;
  const int lane = t & 31;
  const int bbase = (bidx & ~(kN - 1)) * kD;   // b * N * D
  const float b2v = b2[0];
  const float* hjb = hj + bbase;

  // Each tile = 8 rows x 256 f32 = 8KB; 256 threads x 2 b128 issues.
  const uint32_t lds0 = (uint32_t)(uintptr_t)(&hjbuf[0][0][0]) + t * 16u;
  const uint32_t lds1 = (uint32_t)(uintptr_t)(&hjbuf[1][0][0]) + t * 16u;

  // Prime the pipeline with tile 0.
  async_load_b128(lds0, hjb + t * 4);
  async_load_b128(lds0 + 4096u, hjb + 1024 + t * 4);

  for (int mt = 0; mt < kN / 8; ++mt) {
    if (mt + 1 < kN / 8) {
      const float* g = hjb + (mt + 1) * 8 * kD + t * 4;
      const uint32_t l = ((mt + 1) & 1) ? lds1 : lds0;
      async_load_b128(l, g);
      async_load_b128(l + 4096u, g + 1024);
      asm volatile("s_wait_asynccnt 0x2" ::: "memory");  // tile mt landed
    } else {
      asm volatile("s_wait_asynccnt 0x0" ::: "memory");
    }
    __syncthreads();                     // tile mt visible to all waves

    const int m = mt * 8 + wave;
    const float* hjrow = &hjbuf[mt & 1][wave][0];
    float p = 0.0f;
#pragma unroll
    for (int kk = 0; kk < kD / 32; ++kk) {
      const int d = lane + kk * 32;
      const float v = hi_s[d] + hjrow[d];
      p += fmaxf(v, 0.0f) * w2_s[d];
    }
#pragma unroll
    for (int off = 16; off > 0; off >>= 1) p += __shfl_xor(p, off, 32);
    if (lane == 0) sc_s[m] = (m == n) ? 0.0f : (p + b2v);
    __syncthreads();                     // done reading buf before re-issue
  }

  // Softmax over the 256 scores in LDS.
  const float v = sc_s[t];
  float mx = v;
#pragma unroll
  for (int off = 16; off > 0; off >>= 1) mx = fmaxf(mx, __shfl_xor(mx, off, 32));
  if (lane == 0) red_s[wave] = mx;
  __syncthreads();
  float bmax = red_s[0];
#pragma unroll
  for (int i = 1; i < 8; ++i) bmax = fmaxf(bmax, red_s[i]);

  const float e = __expf(v - bmax);
  float s = e;
#pragma unroll
  for (int off = 16; off > 0; off >>= 1) s += __shfl_xor(s, off, 32);
  if (lane == 0) red_s[8 + wave] = s;
  __syncthreads();
  float bsum = 0.0f;
#pragma unroll
  for (int i = 0; i < 8; ++i) bsum += red_s[8 + i];

  wout[bidx * kN + t] = e / bsum;
}

// ---------------------------------------------------------------------------
// Kernel 3: ctx[b] = weights[b] @ x[b]  (8 batched 256^3 GEMMs, WMMA)
// One wave per 16x64 strip: 4 accumulators share the A fragment.
// 512 waves -> 64 blocks of 8 waves.
// ---------------------------------------------------------------------------
__global__ void k3_ctx(const float* __restrict__ wts, const float* __restrict__ x,
                       float* __restrict__ ctx) {
  const int wid  = blockIdx.x * (blockDim.x >> 5) + (threadIdx.x >> 5);
  const int lane = threadIdx.x & 31;
  const int l16  = lane & 15;
  const int half = lane >> 4;
  const int b   = wid >> 6;              // 64 strips per batch
  const int t   = wid & 63;
  const int mt  = t >> 2;
  const int nt4 = t & 3;

  const float* A  = wts + b * kN * kN + mt * 16 * kN;
  const float* Bm = x + b * kN * kD + nt4 * 64;
  float* out = ctx + b * kN * kD + mt * 16 * kD + nt4 * 64;

  v8f acc[4] = {};
  for (int k = 0; k < kN; k += 4) {
    v2f a;
    const float* ap = A + l16 * kN + k + half * 2;
    a.x = ap[0]; a.y = ap[1];
    const float* brow = Bm + (k + half * 2) * kD + l16;
#pragma unroll
    for (int j = 0; j < 4; ++j) {
      v2f bf;
      bf.x = brow[j * 16];
      bf.y = brow[j * 16 + kD];
      acc[j] = __builtin_amdgcn_wmma_f32_16x16x4_f32(false, a, false, bf,
                                                     (short)0, acc[j], false, false);
    }
  }
#pragma unroll
  for (int j = 0; j < 4; ++j)
#pragma unroll
    for (int v = 0; v < 8; ++v)
      out[(v + half * 8) * kD + j * 16 + l16] = acc[j][v];
}

// ---------------------------------------------------------------------------
// Kernel 4: proj = ctx @ Wp + bp, then LayerNorm(gamma,beta) + residual x.
// Block computes a full 16-row x 256-col stripe: 8 waves x 2 WMMA tiles each
// (shared A fragment), stage in LDS, then per-row LN reduction in-block.
// ---------------------------------------------------------------------------
__global__ void k4_proj_ln(const float* __restrict__ ctx, const float* __restrict__ Wp,
                           const float* __restrict__ bias, const float* __restrict__ gamma,
                           const float* __restrict__ beta, const float* __restrict__ x,
                           float* __restrict__ out) {
  __shared__ float proj_s[16][kD];
  const int wave = threadIdx.x >> 5;
  const int lane = threadIdx.x & 31;
  const int l16  = lane & 15;
  const int half = lane >> 4;
  const int r0 = blockIdx.x * 16;
  const int c0 = wave * 32;
  const int c1 = c0 + 16;

  const float* A = ctx + r0 * kD;
  v8f acc0 = {}, acc1 = {};
  for (int k = 0; k < kD; k += 4) {
    v2f a, b0, b1f;
    const float* ap = A + l16 * kD + k + half * 2;
    a.x = ap[0]; a.y = ap[1];
    const float* bp0 = Wp + (k + half * 2) * kD + c0 + l16;
    b0.x  = bp0[0];  b0.y  = bp0[kD];
    b1f.x = bp0[16]; b1f.y = bp0[16 + kD];
    acc0 = __builtin_amdgcn_wmma_f32_16x16x4_f32(false, a, false, b0,
                                                 (short)0, acc0, false, false);
    acc1 = __builtin_amdgcn_wmma_f32_16x16x4_f32(false, a, false, b1f,
                                                 (short)0, acc1, false, false);
  }
#pragma unroll
  for (int v = 0; v < 8; ++v) {
    proj_s[v + half * 8][c0 + l16] = acc0[v] + bias[c0 + l16];
    proj_s[v + half * 8][c1 + l16] = acc1[v] + bias[c1 + l16];
  }
  __syncthreads();

  // LayerNorm: wave handles rows 2*wave and 2*wave+1.
#pragma unroll
  for (int rr = 0; rr < 2; ++rr) {
    const int r = wave * 2 + rr;
    float s = 0.0f;
#pragma unroll
    for (int kk = 0; kk < kD / 32; ++kk) s += proj_s[r][lane + kk * 32];
#pragma unroll
    for (int off = 16; off > 0; off >>= 1) s += __shfl_xor(s, off, 32);
    const float mu = s * (1.0f / kD);

    float vs = 0.0f;
#pragma unroll
    for (int kk = 0; kk < kD / 32; ++kk) {
      const float dv = proj_s[r][lane + kk * 32] - mu;
      vs += dv * dv;
    }
#pragma unroll
    for (int off = 16; off > 0; off >>= 1) vs += __shfl_xor(vs, off, 32);
    const float rstd = rsqrtf(vs * (1.0f / kD) + kLnEps);

    const int grow = r0 + r;
#pragma unroll
    for (int kk = 0; kk < kD / 32; ++kk) {
      const int d = lane + kk * 32;
      out[grow * kD + d] = x[grow * kD + d] +
          (proj_s[r][d] - mu) * rstd * gamma[d] + beta[d];
    }
  }
}

// ---------------------------------------------------------------------------
extern "C" void kernel_launch(void* const* d_in, const int* in_sizes, int n_in,
                              void* d_out, int out_size, void* d_ws, size_t ws_size,
                              hipStream_t stream) {
  (void)in_sizes; (void)n_in; (void)out_size; (void)ws_size;
  const float* x   = (const float*)d_in[0];   // [B,N,D]
  const float* W1  = (const float*)d_in[1];   // [2D,D]
  const float* b1  = (const float*)d_in[2];   // [D]
  const float* W2  = (const float*)d_in[3];   // [D,1]
  const float* b2  = (const float*)d_in[4];   // [1]
  const float* Wp  = (const float*)d_in[5];   // [D,D]
  const float* bp  = (const float*)d_in[6];   // [D]
  const float* gm  = (const float*)d_in[7];   // [D]
  const float* bt  = (const float*)d_in[8];   // [D]
  float* out = (float*)d_out;

  float* ws  = (float*)d_ws;
  float* hi  = ws;                    // [2048,256]
  float* hj  = ws + 1 * kRows * kD;   // [2048,256]
  float* wts = ws + 2 * kRows * kD;   // [B,256,256] softmax weights
  float* ctx = ws + 3 * kRows * kD;   // [2048,256]

  k1_hproj<<<128, 256, 0, stream>>>(x, W1, b1, hi, hj);
  k2_scores_softmax<<<kRows, 256, 0, stream>>>(hi, hj, W2, b2, wts);
  k3_ctx<<<64, 256, 0, stream>>>(wts, x, ctx);
  k4_proj_ln<<<128, 256, 0, stream>>>(ctx, Wp, bp, gm, bt, x, out);
}